// ARMPSShare_13400297963988
// MI455X (gfx1250) — compile-verified
//
#include <hip/hip_runtime.h>

// ============================================================================
// ARMPS chain on MI455X (gfx1250): batch-16-per-wave sequential scan where each
// step is 4 candidate GEMMs (16x16x16 f32) done as 4x4 chained
// V_WMMA_F32_16X16X4_F32, with per-lane (per-batch) candidate selection.
//
// State leftT kept in WMMA C layout (lane = batch col, VGPR = row).
// C -> B operand relayout = 8x ds_swizzle(SWAPX16) + 8x cndmask.
// Logits tmp[b,i] = row 0 of candidate Y_i => C component 0, lanes 0..15.
// Candidate selection done via one-hot weighted FMA (branch-free, keeps EXEC
// all-ones and dual-issues as v_dual_fmac).
// ============================================================================

typedef __attribute__((ext_vector_type(2))) float v2f;
typedef __attribute__((ext_vector_type(8))) float v8f;

#define N_POS 128
#define DDIM  16
#define FDIM  4

// ---------------------------------------------------------------------------
// Prep: rearrange tensors (+ identity bias) into WMMA A-operand lane layout.
// A operand (per step n, candidate i) is Aop[r,l] = tensors[n,l,r,i] + (l==r).
// A-layout (f32 16x16 via 4 K=4 slices): VGPR j (j=2g+d):
//   lane<16 : M=lane,    K = 4g + d
//   lane>=16: M=lane-16, K = 4g + d + 2
// ws layout: ws[((n*4 + i)*32 + lane)*8 + j]   (128*4*32*8 = 131072 floats)
// ---------------------------------------------------------------------------
__global__ void armps_prep_kernel(const float* __restrict__ tensors,
                                  float* __restrict__ wsA) {
    int idx = blockIdx.x * blockDim.x + threadIdx.x;
    if (idx >= N_POS * FDIM * 32 * 8) return;
    int j    = idx & 7;
    int lane = (idx >> 3) & 31;
    int i    = (idx >> 8) & 3;
    int n    = idx >> 10;
    int h = lane >> 4;          // lane half
    int r = lane & 15;          // output row (M)
    int g = j >> 1;
    int d = j & 1;
    int l = 4 * g + 2 * h + d;  // K index
    float v = tensors[((size_t)(n * DDIM + l) * DDIM + r) * FDIM + i]
              + ((l == r) ? 1.0f : 0.0f);
    wsA[idx] = v;
}

// swap lane halves (lane ^ 16) via LDS swizzle unit (no LDS memory touched)
__device__ __forceinline__ float swz16(float x) {
    // group-of-32: and=0x1f, or=0, xor=0x10  => SWAPX16
    return __int_as_float(__builtin_amdgcn_ds_swizzle(__float_as_int(x), 0x401F));
}

#define WM(Y, AA, BB) \
    Y = __builtin_amdgcn_wmma_f32_16x16x4_f32(false, AA, false, BB, (short)0, Y, false, false)

__global__ void __launch_bounds__(256)
armps_chain_kernel(const int* __restrict__ data,
                   const float* __restrict__ tensors,
                   const float* __restrict__ wsA,
                   float* __restrict__ out) {
    const int lane   = threadIdx.x & 31;
    const int wave   = (int)((blockIdx.x * blockDim.x + threadIdx.x) >> 5);
    const int base   = wave * 16;          // 16 batch rows per wave
    const int r_lane = lane & 15;          // batch column this lane owns
    const bool lo    = lane < 16;

    const int* rowptr = data + (size_t)(base + r_lane) * N_POS;

    // f values for n = 0..3
    int4 dv = *(const int4*)(rowptr);
    const int f0 = dv.x;

    // n = 0 contribution: log0 = log_softmax(tensors[0,0,0,:] + 1)[f0]
    float q0 = tensors[0] + 1.f, q1 = tensors[1] + 1.f;
    float q2 = tensors[2] + 1.f, q3 = tensors[3] + 1.f;
    float m0 = fmaxf(fmaxf(q0, q1), fmaxf(q2, q3));
    float z0 = __expf(q0 - m0) + __expf(q1 - m0) + __expf(q2 - m0) + __expf(q3 - m0);
    float tq = (f0 == 0) ? q0 : (f0 == 1) ? q1 : (f0 == 2) ? q2 : q3;
    float acc = tq - m0 - __logf(z0);

    // left0 in C layout: Lc[v] holds leftT[row = v + 8*half, b = lane&15]
    // left0[r,b] = tensors[0,0,r,f0_b] + (r==0)
    float Lc[8];
#pragma unroll
    for (int v = 0; v < 8; ++v) {
        int r = v + (lo ? 0 : 8);
        Lc[v] = tensors[r * FDIM + f0] + ((r == 0) ? 1.f : 0.f);
    }

#pragma unroll 1
    for (int nb = 0; nb < N_POS; nb += 4) {
        if (nb) dv = *(const int4*)(rowptr + nb);   // uniform branch
#pragma unroll
        for (int k = 0; k < 4; ++k) {
            const int n = nb + k;
            if (n == 0) continue;                    // compile-time under unroll
            const int f = (k == 0) ? dv.x : (k == 1) ? dv.y : (k == 2) ? dv.z : dv.w;

            // --- load 4 candidate A operands (pre-swizzled, bias folded) ---
            const float* wb = wsA + (size_t)n * 1024 + lane * 8;
            float4 a00 = *(const float4*)(wb +   0);
            float4 a01 = *(const float4*)(wb +   4);
            float4 a10 = *(const float4*)(wb + 256);
            float4 a11 = *(const float4*)(wb + 260);
            float4 a20 = *(const float4*)(wb + 512);
            float4 a21 = *(const float4*)(wb + 516);
            float4 a30 = *(const float4*)(wb + 768);
            float4 a31 = *(const float4*)(wb + 772);
            if (n + 1 < N_POS)                        // uniform
                __builtin_prefetch(wsA + (size_t)(n + 1) * 1024 + lane * 8, 0, 1);

            // --- C-layout -> B-operand relayout (half-wave swap + select) ---
            float s0 = swz16(Lc[0]), s1 = swz16(Lc[1]);
            float s2 = swz16(Lc[2]), s3 = swz16(Lc[3]);
            float s4 = swz16(Lc[4]), s5 = swz16(Lc[5]);
            float s6 = swz16(Lc[6]), s7 = swz16(Lc[7]);
            float b0 = lo ? Lc[0] : s2;   // g0: rows 0,1 | 2,3
            float b1 = lo ? Lc[1] : s3;
            float b2 = lo ? Lc[4] : s6;   // g1: rows 4,5 | 6,7
            float b3 = lo ? Lc[5] : s7;
            float b4 = lo ? s0 : Lc[2];   // g2: rows 8,9 | 10,11
            float b5 = lo ? s1 : Lc[3];
            float b6 = lo ? s4 : Lc[6];   // g3: rows 12,13 | 14,15
            float b7 = lo ? s5 : Lc[7];
            v2f B0 = {b0, b1}, B1 = {b2, b3}, B2 = {b4, b5}, B3 = {b6, b7};

            // --- 4 candidate GEMMs: Y_i = A_n(:,:,i)^T @ leftT  (K=16) ---
            v8f Y0 = {}, Y1 = {}, Y2 = {}, Y3 = {};
            {
                v2f A0 = {a00.x, a00.y}, A1 = {a00.z, a00.w};
                v2f A2 = {a01.x, a01.y}, A3 = {a01.z, a01.w};
                WM(Y0, A0, B0); WM(Y0, A1, B1); WM(Y0, A2, B2); WM(Y0, A3, B3);
            }
            {
                v2f A0 = {a10.x, a10.y}, A1 = {a10.z, a10.w};
                v2f A2 = {a11.x, a11.y}, A3 = {a11.z, a11.w};
                WM(Y1, A0, B0); WM(Y1, A1, B1); WM(Y1, A2, B2); WM(Y1, A3, B3);
            }
            {
                v2f A0 = {a20.x, a20.y}, A1 = {a20.z, a20.w};
                v2f A2 = {a21.x, a21.y}, A3 = {a21.z, a21.w};
                WM(Y2, A0, B0); WM(Y2, A1, B1); WM(Y2, A2, B2); WM(Y2, A3, B3);
            }
            {
                v2f A0 = {a30.x, a30.y}, A1 = {a30.z, a30.w};
                v2f A2 = {a31.x, a31.y}, A3 = {a31.z, a31.w};
                WM(Y3, A0, B0); WM(Y3, A1, B1); WM(Y3, A2, B2); WM(Y3, A3, B3);
            }

            // --- one-hot selection weights (branch-free; values are finite) ---
            const float w0 = (f == 0) ? 1.f : 0.f;
            const float w1 = (f == 1) ? 1.f : 0.f;
            const float w2 = (f == 2) ? 1.f : 0.f;
            const float w3 = (f == 3) ? 1.f : 0.f;

            // --- logits: tmp[b,i] = Y_i row 0 = C component 0 (lanes 0..15) ---
            float t0 = Y0[0], t1 = Y1[0], t2 = Y2[0], t3 = Y3[0];
            float mm = fmaxf(fmaxf(t0, t1), fmaxf(t2, t3));
            float zz = __expf(t0 - mm) + __expf(t1 - mm) +
                       __expf(t2 - mm) + __expf(t3 - mm);
            float tf = fmaf(w0, t0, fmaf(w1, t1, fmaf(w2, t2, w3 * t3)));
            acc += tf - mm - __logf(zz);

            // --- per-batch (per-lane) candidate select via weighted FMA ---
#pragma unroll
            for (int v = 0; v < 8; ++v)
                Lc[v] = fmaf(w0, Y0[v],
                        fmaf(w1, Y1[v],
                        fmaf(w2, Y2[v], w3 * Y3[v])));
        }
    }

    if (lane < 16) out[base + r_lane] = acc;
}

extern "C" void kernel_launch(void* const* d_in, const int* in_sizes, int n_in,
                              void* d_out, int out_size, void* d_ws, size_t ws_size,
                              hipStream_t stream) {
    const int*   data    = (const int*)d_in[0];     // (32768, 128) int
    const float* tensors = (const float*)d_in[1];   // (128, 16, 16, 4) f32
    float*       out     = (float*)d_out;           // (32768,) f32
    float*       wsA     = (float*)d_ws;            // 512 KB A-operand cache

    // 128*4*32*8 = 131072 prep elements
    armps_prep_kernel<<<(131072 + 255) / 256, 256, 0, stream>>>(tensors, wsA);

    // 32768 batches / 16 per wave = 2048 waves = 65536 threads
    armps_chain_kernel<<<65536 / 256, 256, 0, stream>>>(data, tensors, wsA, out);
}